// FlowNetC_35493609734315
// MI455X (gfx1250) — compile-verified
//
#include <hip/hip_runtime.h>

typedef __attribute__((ext_vector_type(2))) float v2f;
typedef __attribute__((ext_vector_type(8))) float v8f;

namespace {
constexpr int CH   = 256;          // channels (K)
constexpr int H    = 48;
constexpr int W    = 64;
constexpr int HW   = H * W;
constexpr int NOFF = 21;           // 21 displacements per axis (stride 2 over +-20)
constexpr int PAD  = 20;
}

// One block: one (row h, 16 same-parity pixels) A-tile.
// 8 waves round-robin the 21 vertical shifts oy.
// Per oy: 3 Gram tiles C[m][n] = sum_c f1[c,h,wbase+2m] * f2[c,h+oy, wbase-20+2(16t+n)]
// computed with V_WMMA_F32_16X16X4_F32 over a 64-step K loop.
__global__ __launch_bounds__(256)
void flownetc_corr_wmma(const float* __restrict__ f1,
                        const float* __restrict__ f2,
                        float* __restrict__ out) {
    // A staged as channel-pairs so each WMMA k-step is a single ds_load_b64:
    // lane needs channels (4kk + 2*khalf, 4kk + 2*khalf + 1) of pixel m.
    __shared__ float2 Alds[CH / 2][16];

    const int tid   = threadIdx.x;
    const int h     = blockIdx.x >> 2;
    const int wt    = blockIdx.x & 3;
    const int wbase = ((wt >> 1) << 5) + (wt & 1);   // 0, 1, 32, 33

    // Cooperative stage of A = f1[:, h, wbase + 2m], m = 0..15  (16 KB LDS)
    for (int idx = tid; idx < (CH / 2) * 16; idx += 256) {
        const int cp = idx >> 4;           // channel pair
        const int m  = idx & 15;
        const int g  = (2 * cp) * HW + h * W + wbase + 2 * m;
        Alds[cp][m] = make_float2(f1[g], f1[g + HW]);
    }
    __syncthreads();

    const int wave  = tid >> 5;            // wave32
    const int lane  = tid & 31;
    const int mn    = lane & 15;           // M for A, N for B/C
    const int khalf = lane >> 4;           // which K pair of the 4-K group

    // Per-lane B column for each of the 3 Gram tiles: clamped index + validity
    // mask. Mask depends only on (n, t), so it is applied to the accumulator
    // columns AFTER the k-loop -> inner loop has no divergence (EXEC all-1s).
    int   jc[3];
    float msk[3];
#pragma unroll
    for (int t = 0; t < 3; ++t) {
        const int j = wbase - PAD + 32 * t + 2 * mn;
        msk[t] = (j >= 0 && j < W) ? 1.0f : 0.0f;
        jc[t]  = j < 0 ? 0 : (j > W - 1 ? W - 1 : j);
    }

    for (int oyi = wave; oyi < NOFF; oyi += 8) {
        const int r2 = h + 2 * oyi - PAD;  // f2 row for this vertical shift

        // Prefetch next oy row for this wave (global_prefetch_b8).
        const int r2n = h + 2 * (oyi + 8) - PAD;
        if (oyi + 8 < NOFF && r2n >= 0 && r2n < H)
            __builtin_prefetch(f2 + (size_t)(2 * khalf) * HW + r2n * W + jc[0], 0, 0);

        v8f acc[3];
#pragma unroll
        for (int t = 0; t < 3; ++t)
            acc[t] = (v8f){0.f, 0.f, 0.f, 0.f, 0.f, 0.f, 0.f, 0.f};

        if (r2 >= 0 && r2 < H) {           // wave-uniform branch
            const float*  p  = f2 + (size_t)(2 * khalf) * HW + r2 * W;
            const float2* ap = &Alds[khalf][mn];
#pragma unroll 8
            for (int kk = 0; kk < CH / 4; ++kk) {
                const float2 av = ap[kk * 32];   // Alds[2*kk + khalf][mn]
                v2f a; a.x = av.x; a.y = av.y;
                v2f b0, b1, b2;
                b0.x = p[jc[0]]; b0.y = p[HW + jc[0]];
                b1.x = p[jc[1]]; b1.y = p[HW + jc[1]];
                b2.x = p[jc[2]]; b2.y = p[HW + jc[2]];
                acc[0] = __builtin_amdgcn_wmma_f32_16x16x4_f32(
                             false, a, false, b0, (short)0, acc[0], false, false);
                acc[1] = __builtin_amdgcn_wmma_f32_16x16x4_f32(
                             false, a, false, b1, (short)0, acc[1], false, false);
                acc[2] = __builtin_amdgcn_wmma_f32_16x16x4_f32(
                             false, a, false, b2, (short)0, acc[2], false, false);
                p += 4 * HW;               // advance 4 channels
            }
        }
        // else: accumulators stay zero -> zero-padding rows still get written.

        // Zero the out-of-image B columns.
#pragma unroll
        for (int t = 0; t < 3; ++t)
#pragma unroll
            for (int r = 0; r < 8; ++r)
                acc[t][r] *= msk[t];

        // Scatter the Gram-band diagonals: C VGPR r holds M = r + 8*khalf,
        // N = lane&15; ox index = 16t + n - m, valid in [0, 21). Each output
        // element is produced by exactly one (lane, t, r).
#pragma unroll
        for (int t = 0; t < 3; ++t) {
#pragma unroll
            for (int r = 0; r < 8; ++r) {
                const int mo  = r + 8 * khalf;
                const int oxi = 16 * t + mn - mo;
                if (oxi >= 0 && oxi < NOFF) {
                    const int wo = wbase + 2 * mo;
                    out[((size_t)(oyi * NOFF + oxi) * H + h) * W + wo] = acc[t][r];
                }
            }
        }
    }
}

extern "C" void kernel_launch(void* const* d_in, const int* in_sizes, int n_in,
                              void* d_out, int out_size, void* d_ws, size_t ws_size,
                              hipStream_t stream) {
    (void)in_sizes; (void)n_in; (void)out_size; (void)d_ws; (void)ws_size;
    const float* f1  = (const float*)d_in[0];
    const float* f2  = (const float*)d_in[1];
    float*       out = (float*)d_out;

    dim3 grid(H * 4);   // 48 rows x 4 parity tiles = 192 blocks
    dim3 block(256);    // 8 wave32 waves
    flownetc_corr_wmma<<<grid, block, 0, stream>>>(f1, f2, out);
}